// LearnedPosMap_19610820673580
// MI455X (gfx1250) — compile-verified
//
#include <hip/hip_runtime.h>

typedef __attribute__((ext_vector_type(2))) float v2f;
typedef __attribute__((ext_vector_type(8))) float v8f;

#define SEQn   64
#define DIMC   384
#define NGRP   4
#define TABLEn 225
#define NWS    256                    // BLOCKS*GAMMA
#define XT_STRIDE 65                  // xT[c][n] row stride: bank = (c+k)%64 -> conflict-free B frags
#define A_STRIDE  68                  // A[s][m][n] row stride: conflict-free A frags
#define XT_DW  (DIMC * XT_STRIDE)     // 24960 dwords
#define A_MAT  (SEQn * A_STRIDE)      // 4352 dwords per s
#define A_OFF  XT_DW
#define LDS_DW (XT_DW + NGRP * A_MAT) // 42368 dwords = 169472 bytes

// ---------------- pre-pass: BT[w][s][idx] = bias_table[idx][s*64+w] ----------------
__global__ void lpm_bt_transpose(const float* __restrict__ bias_table,
                                 float* __restrict__ BT) {
    int i = blockIdx.x * 256 + threadIdx.x;      // 0 .. 57599
    if (i < NWS * TABLEn) {
        int ws  = i / TABLEn;                    // w*4 + s
        int idx = i - ws * TABLEn;
        int s   = ws & 3;
        int w   = ws >> 2;
        BT[i] = bias_table[idx * NWS + s * 64 + w];
    }
}

// ---------------- main: per (b,w) block, 16 waves -> (s, m-tile) ----------------
__global__ __launch_bounds__(512)
void lpm_main(const float* __restrict__ x,
              const float* __restrict__ BT,
              const float* __restrict__ token_bias,
              const int*   __restrict__ rel_index,
              float* __restrict__ out) {
    extern __shared__ float lds[];
    const int tid = threadIdx.x;
    const int blk = blockIdx.x;
    const int w = blk >> 5;                      // window 0..63 (nearby blocks share BT rows)
    const int b = blk & 31;                      // batch  0..31

    // ---- Phase A: stage xT[c][n] (transposed) and A[s][m][n] into LDS ----
    const float4* xsrc4 = (const float4*)(x + (size_t)(b * 64 + w) * (SEQn * DIMC));
    #pragma unroll
    for (int it = 0; it < 12; ++it) {
        int i4 = tid + it * 512;                 // 0..6143 float4's
        int n  = i4 / 96;
        int cq = i4 - n * 96;                    // channel quad
        float4 v = xsrc4[i4];                    // coalesced b128
        int base = cq * 4 * XT_STRIDE + n;
        lds[base]                 = v.x;
        lds[base + XT_STRIDE]     = v.y;
        lds[base + 2 * XT_STRIDE] = v.z;
        lds[base + 3 * XT_STRIDE] = v.w;
    }
    const float* BTw = BT + (w * 4) * TABLEn;    // 4 x 225 floats: L0-resident gather set
    #pragma unroll
    for (int it = 0; it < 32; ++it) {
        int e  = tid + it * 512;                 // 0..16383
        int s  = e >> 12;
        int mn = e & 4095;
        int m  = mn >> 6;
        int n  = mn & 63;
        lds[A_OFF + s * A_MAT + m * A_STRIDE + n] = BTw[s * TABLEn + rel_index[mn]];
    }

    // wave roles
    const int lane  = tid & 31;
    const int wv    = tid >> 5;                  // 0..15
    const int s     = wv & 3;                    // channel group
    const int mt    = wv >> 2;                   // m-tile 0..3
    const int lhalf = lane >> 4;                 // 0/1
    const int j     = lane & 15;

    // token bias: depends only on (w,s,m); uniform per half-wave per acc VGPR
    float biasv[8];
    {
        const float* tb = token_bias + ((w * 4 + s) * 64 + mt * 16 + 8 * lhalf);
        #pragma unroll
        for (int r = 0; r < 8; ++r) biasv[r] = tb[r];
    }
    __syncthreads();

    // ---- Phase B: K=64 loop of v_wmma_f32_16x16x4_f32, 6 N-tiles per wave ----
    v8f acc[6];
    #pragma unroll
    for (int nt = 0; nt < 6; ++nt) acc[nt] = (v8f)(0.0f);

    // A frag: lane -> A[s][mt*16+j][k0 + 2*lhalf + {0,1}]  (even base, conflict-free)
    const int a_base = A_OFF + s * A_MAT + (mt * 16 + j) * A_STRIDE + 2 * lhalf;
    // B frag: lane -> xT[(nt*16+j)*4+s][k0 + 2*lhalf + {0,1}] (conflict-free b32 pairs)
    const int b_base = (j * 4 + s) * XT_STRIDE + 2 * lhalf;

    #pragma unroll 4
    for (int k0 = 0; k0 < 64; k0 += 4) {
        v2f a = *(const v2f*)&lds[a_base + k0];
        #pragma unroll
        for (int nt = 0; nt < 6; ++nt) {
            int ba = b_base + nt * (64 * XT_STRIDE) + k0;
            v2f bb;
            bb.x = lds[ba];          // fuses to ds_load_2addr_b32 offset1:1
            bb.y = lds[ba + 1];
            acc[nt] = __builtin_amdgcn_wmma_f32_16x16x4_f32(
                false, a, false, bb, (short)0, acc[nt], false, false);
        }
    }
    __syncthreads();

    // ---- Phase C: acc + bias -> LDS out[c][m] (reuse xT region) ----
    const int mbase = mt * 16 + 8 * lhalf;
    #pragma unroll
    for (int nt = 0; nt < 6; ++nt) {
        int c = (nt * 16 + j) * 4 + s;
        #pragma unroll
        for (int r = 0; r < 8; ++r)
            lds[c * XT_STRIDE + mbase + r] = acc[nt][r] + biasv[r];
    }
    __syncthreads();

    // ---- Phase D: coalesced b128 store ----
    float4* odst4 = (float4*)(out + (size_t)(b * 64 + w) * (SEQn * DIMC));
    #pragma unroll
    for (int it = 0; it < 12; ++it) {
        int i4 = tid + it * 512;
        int m  = i4 / 96;
        int cq = i4 - m * 96;
        int base = cq * 4 * XT_STRIDE + m;
        float4 v;
        v.x = lds[base];
        v.y = lds[base + XT_STRIDE];
        v.z = lds[base + 2 * XT_STRIDE];
        v.w = lds[base + 3 * XT_STRIDE];
        odst4[i4] = v;
    }
}

extern "C" void kernel_launch(void* const* d_in, const int* in_sizes, int n_in,
                              void* d_out, int out_size, void* d_ws, size_t ws_size,
                              hipStream_t stream) {
    const float* x          = (const float*)d_in[0];
    const float* bias_table = (const float*)d_in[1];
    const float* token_bias = (const float*)d_in[2];
    const int*   rel_index  = (const int*)d_in[3];
    float* out = (float*)d_out;
    float* BT  = (float*)d_ws;                  // 57600 floats = 230 KB scratch

    (void)hipFuncSetAttribute((const void*)lpm_main,
                              hipFuncAttributeMaxDynamicSharedMemorySize,
                              LDS_DW * (int)sizeof(float));

    lpm_bt_transpose<<<(NWS * TABLEn + 255) / 256, 256, 0, stream>>>(bias_table, BT);
    lpm_main<<<2048, 512, LDS_DW * sizeof(float), stream>>>(x, BT, token_bias,
                                                            rel_index, out);
}